// YoloDetector_51548197486703
// MI455X (gfx1250) — compile-verified
//
#include <hip/hip_runtime.h>
#include <stdint.h>

// YOLO post-process: x (B, 30, 7, 7) f32 -> boxes (B,98,4), scores (B,98), idxs (B,98)
// Memory-bound (~135 MB traffic, ~6 us at 23.3 TB/s). CDNA5 path: async global->LDS
// staging (ASYNCcnt) + wave32 blocks, register compute, 128-bit coalesced stores.

#define NCLS 20
#define CELLS 49            // 7*7
#define CH 30               // 5*NBOX + NCLS
#define IMG_FLOATS 1470     // CH * CELLS
#define IPB 4               // images per block (4*1470 floats = 1470 float4 chunks)
#define TPI 64              // threads per image
#define BLOCK (IPB * TPI)   // 256 threads = 8 waves (wave32)
#define CHUNKS ((IPB * IMG_FLOATS) / 4)  // 1470 b128 chunks per block

__device__ __forceinline__ void async_copy_b128(unsigned lds_off, unsigned long long gaddr) {
    // CDNA5: async DMA global -> LDS, tracked by ASYNCcnt (no VGPR return traffic).
    asm volatile("global_load_async_to_lds_b128 %0, %1, off"
                 :: "v"(lds_off), "v"(gaddr) : "memory");
}

__device__ __forceinline__ void async_copy_b32(unsigned lds_off, unsigned long long gaddr) {
    asm volatile("global_load_async_to_lds_b32 %0, %1, off"
                 :: "v"(lds_off), "v"(gaddr) : "memory");
}

__device__ __forceinline__ void wait_async_done() {
    asm volatile("s_wait_asynccnt 0" ::: "memory");
}

__global__ __launch_bounds__(BLOCK) void yolo_post_kernel(
    const float* __restrict__ x,
    float* __restrict__ boxes,   // (B, 98, 4)
    float* __restrict__ scores,  // (B, 98)
    float* __restrict__ idxs,    // (B, 98)
    int n_img)
{
    __shared__ float smem[IPB * IMG_FLOATS];

    const int tid  = threadIdx.x;
    const int img0 = blockIdx.x * IPB;
    if (img0 >= n_img) return;

    // Flat->LDS aperture: low 32 bits of the generic pointer are the LDS byte
    // offset relative to the wave's LDS base (ISA 10.2), which is exactly what
    // the async-to-LDS instruction consumes in its VDST operand.
    const unsigned lds_base = (unsigned)(uintptr_t)(&smem[0]);
    const unsigned long long gbase =
        (unsigned long long)(uintptr_t)x + (unsigned long long)img0 * (IMG_FLOATS * 4ull);

    const int nimg_blk = (n_img - img0 < IPB) ? (n_img - img0) : IPB;
    if (nimg_blk == IPB) {
        // 4 images = 5880 floats = 1470 aligned 16B chunks; fully coalesced B128 DMA.
        #pragma unroll 2
        for (int c = tid; c < CHUNKS; c += BLOCK) {
            async_copy_b128(lds_base + (unsigned)c * 16u, gbase + (unsigned long long)c * 16u);
        }
    } else {
        const int nfloats = nimg_blk * IMG_FLOATS;
        for (int f = tid; f < nfloats; f += BLOCK) {
            async_copy_b32(lds_base + (unsigned)f * 4u, gbase + (unsigned long long)f * 4u);
        }
    }
    wait_async_done();   // this wave's async DMAs landed in LDS
    __syncthreads();     // all waves' DMAs visible work-group wide

    const int li   = tid / TPI;       // image slot within block
    const int lane = tid % TPI;       // cell id
    const int img  = img0 + li;
    if (lane >= CELLS || img >= n_img) return;

    const float* __restrict__ p = &smem[li * IMG_FLOATS];
    const int   s  = lane;
    const float gx = (float)(s % 7);
    const float gy = (float)(s / 7);

    // Confidence-weighted class max/argmax for both boxes (first-index tiebreak,
    // strict '>' matches jnp.argmax semantics).
    const float conf0 = p[4 * CELLS + s];
    const float conf1 = p[9 * CELLS + s];
    float c0    = p[(5 * 2 + 0 + 0) * CELLS + s]; // class 0 value (channel 10)
    float best0 = c0 * conf0;  int i0 = 0;
    float best1 = c0 * conf1;  int i1 = 0;
    #pragma unroll
    for (int c = 1; c < NCLS; ++c) {
        const float cv = p[(10 + c) * CELLS + s];
        const float w0 = cv * conf0;
        const float w1 = cv * conf1;
        if (w0 > best0) { best0 = w0; i0 = c; }
        if (w1 > best1) { best1 = w1; i1 = c; }
    }

    // Box decode: center = (det_xy + grid)/7 ; half = det_wh/2 ; [xmin,ymin,xmax,ymax]
    float4 b0, b1;
    {
        const float cx = (p[0 * CELLS + s] + gx) / 7.0f;
        const float cy = (p[1 * CELLS + s] + gy) / 7.0f;
        const float hw = p[2 * CELLS + s] * 0.5f;
        const float hh = p[3 * CELLS + s] * 0.5f;
        b0 = make_float4(cx - hw, cy - hh, cx + hw, cy + hh);
    }
    {
        const float cx = (p[5 * CELLS + s] + gx) / 7.0f;
        const float cy = (p[6 * CELLS + s] + gy) / 7.0f;
        const float hw = p[7 * CELLS + s] * 0.5f;
        const float hh = p[8 * CELLS + s] * 0.5f;
        b1 = make_float4(cx - hw, cy - hh, cx + hw, cy + hh);
    }

    // Stores: boxes rows are 16B-aligned -> b128 stores; scores/idxs coalesced b32.
    float4* __restrict__ bx = (float4*)(boxes + (size_t)img * 98 * 4);
    bx[s]          = b0;
    bx[CELLS + s]  = b1;

    const size_t so = (size_t)img * 98;
    scores[so + s]         = best0;
    scores[so + CELLS + s] = best1;
    idxs[so + s]           = (float)i0;
    idxs[so + CELLS + s]   = (float)i1;
}

extern "C" void kernel_launch(void* const* d_in, const int* in_sizes, int n_in,
                              void* d_out, int out_size, void* d_ws, size_t ws_size,
                              hipStream_t stream) {
    (void)n_in; (void)out_size; (void)d_ws; (void)ws_size;
    const float* x = (const float*)d_in[0];
    const int B = in_sizes[0] / (CH * CELLS);   // 16384

    float* out    = (float*)d_out;
    float* boxes  = out;                                   // B*98*4
    float* scores = out + (size_t)B * 98 * 4;              // B*98
    float* idxs   = scores + (size_t)B * 98;               // B*98

    dim3 grid((B + IPB - 1) / IPB);
    dim3 block(BLOCK);
    yolo_post_kernel<<<grid, block, 0, stream>>>(x, boxes, scores, idxs, B);
}